// KGEncoderGAT_31954556682723
// MI455X (gfx1250) — compile-verified
//
#include <hip/hip_runtime.h>
#include <math.h>

#define N_NODES 10000
#define IN_DIM  300
#define HID     256
#define HEADS   4
#define FDIM    (HEADS * HID)   // 1024
#define NEG_SLOPE 0.2f

typedef __attribute__((ext_vector_type(2))) float v2f;
typedef __attribute__((ext_vector_type(8))) float v8f;

// ---------------------------------------------------------------------------
// CSR construction: count -> scan -> fill.  Virtual edge list = E real edges
// followed by N self loops (matches reference's concatenate([edges, loop])).
// ---------------------------------------------------------------------------
__global__ void count_kernel(const long long* __restrict__ ei, int E, int* __restrict__ counts) {
    int e = blockIdx.x * blockDim.x + threadIdx.x;
    int Et = E + N_NODES;
    if (e >= Et) return;
    int dst = (e < E) ? (int)ei[(size_t)E + e] : (e - E);
    atomicAdd(&counts[dst], 1);
}

__global__ void scan_kernel(const int* __restrict__ counts, int* __restrict__ row_ptr,
                            int* __restrict__ cursor, int n) {
    __shared__ int buf[1024];
    __shared__ int s_running;
    int tid = threadIdx.x;
    if (tid == 0) s_running = 0;
    __syncthreads();
    for (int base = 0; base < n; base += 1024) {
        int v = (base + tid < n) ? counts[base + tid] : 0;
        buf[tid] = v;
        __syncthreads();
        for (int off = 1; off < 1024; off <<= 1) {
            int t = (tid >= off) ? buf[tid - off] : 0;
            __syncthreads();
            buf[tid] += t;
            __syncthreads();
        }
        int incl = buf[tid];
        int excl = s_running + incl - v;
        if (base + tid < n) { row_ptr[base + tid] = excl; cursor[base + tid] = excl; }
        __syncthreads();
        if (tid == 1023) s_running += incl;
        __syncthreads();
    }
    if (tid == 0) row_ptr[n] = s_running;
}

__global__ void fill_kernel(const long long* __restrict__ ei, int E,
                            int* __restrict__ cursor, int* __restrict__ col_src) {
    int e = blockIdx.x * blockDim.x + threadIdx.x;
    int Et = E + N_NODES;
    if (e >= Et) return;
    int src, dst;
    if (e < E) { src = (int)ei[e]; dst = (int)ei[(size_t)E + e]; }
    else       { src = dst = e - E; }
    int pos = atomicAdd(&cursor[dst], 1);
    col_src[pos] = src;
}

// ---------------------------------------------------------------------------
// FP32 WMMA GEMM: C[M,N] = A[M,K] * B[K,N], row-major, K % 4 == 0, M % 16 == 0,
// N tiled 64 per block (4 waves * 16 cols).  A tile staged in LDS.
// ---------------------------------------------------------------------------
__global__ void gemm_wmma_f32(const float* __restrict__ A, const float* __restrict__ B,
                              float* __restrict__ C, int K, int N) {
    extern __shared__ float lds_a[];               // 16 x K
    const int tid   = threadIdx.x;                 // 128 threads = 4 waves
    const int wave  = tid >> 5;
    const int lane  = tid & 31;
    const int mbase = blockIdx.x * 16;
    const int nbase = blockIdx.y * 64 + wave * 16;

    for (int idx = tid; idx < 16 * K; idx += 128) {
        int r = idx / K, c = idx - r * K;
        lds_a[idx] = A[(size_t)(mbase + r) * K + c];
    }
    __syncthreads();

    const int half = lane >> 4;                    // 0: K pair {0,1}; 1: {2,3}
    const int lrow = lane & 15;
    v8f acc = {};
    for (int k = 0; k < K; k += 4) {
        int k0 = k + 2 * half;
        v2f a, b;
        a.x = lds_a[lrow * K + k0];
        a.y = lds_a[lrow * K + k0 + 1];
        b.x = B[(size_t)(k0)     * N + nbase + lrow];
        b.y = B[(size_t)(k0 + 1) * N + nbase + lrow];
        acc = __builtin_amdgcn_wmma_f32_16x16x4_f32(false, a, false, b,
                                                    (short)0, acc, false, false);
    }
    // C/D layout: VGPR v, lanes 0-15 -> row v, lanes 16-31 -> row v+8
    #pragma unroll
    for (int v = 0; v < 8; ++v) {
        int row = mbase + v + 8 * half;
        C[(size_t)row * N + nbase + lrow] = acc[v];
    }
}

// ---------------------------------------------------------------------------
// Per-node attention coefficients: a_src[n][h] = <h[n,h,:], att_src[h,:]>
// One block (256 threads) per node.  Thread t owns channels [4t, 4t+3]
// (all within head t>>6) -> single b128 load per thread; per-64-thread
// group reduction produces each head's dot product.
// ---------------------------------------------------------------------------
__global__ void attn_kernel(const float* __restrict__ hfeat,
                            const float* __restrict__ att_src,
                            const float* __restrict__ att_dst,
                            float* __restrict__ a_src, float* __restrict__ a_dst) {
    __shared__ float red[2 * 256];
    const int n = blockIdx.x, t = threadIdx.x;
    const float4 hv = *(const float4*)(hfeat + (size_t)n * FDIM + t * 4);
    const float4 sv = *(const float4*)(att_src + t * 4);
    const float4 dv = *(const float4*)(att_dst + t * 4);
    red[t]       = hv.x * sv.x + hv.y * sv.y + hv.z * sv.z + hv.w * sv.w;
    red[256 + t] = hv.x * dv.x + hv.y * dv.y + hv.z * dv.z + hv.w * dv.w;
    __syncthreads();
    for (int s = 32; s > 0; s >>= 1) {
        if ((t & 63) < s) {
            red[t]       += red[t + s];
            red[256 + t] += red[256 + t + s];
        }
        __syncthreads();
    }
    if (t < HEADS) {
        a_src[n * HEADS + t] = red[t * 64];
        a_dst[n * HEADS + t] = red[256 + t * 64];
    }
}

// ---------------------------------------------------------------------------
// Gather-style edge softmax + aggregation.  One block (256 threads) per
// destination node; streaming (max, sum, weighted-accumulate) softmax, no
// global atomics.  Thread t owns 4 consecutive channels [4t, 4t+3] of head
// t>>6, so each edge message is ONE b128 load per thread.  Head-mean done
// via a small LDS transpose at the end.
// ---------------------------------------------------------------------------
__global__ void aggregate_kernel(const float* __restrict__ hfeat,
                                 const float* __restrict__ a_src,
                                 const float* __restrict__ a_dst,
                                 const int* __restrict__ row_ptr,
                                 const int* __restrict__ col_src,
                                 const float* __restrict__ bias,
                                 float* __restrict__ out, int do_elu) {
    constexpr int CH = 64;
    __shared__ float red[256 * HEADS];      // reductions + final head transpose
    __shared__ int   s_src[CH];
    __shared__ float s_alpha[CH * HEADS];

    const int n = blockIdx.x, tid = threadIdx.x;
    const int start = row_ptr[n];
    const int deg   = row_ptr[n + 1] - start;
    const int head  = tid >> 6;
    const float4* a_src4 = (const float4*)a_src;   // [N][4], 16B-aligned rows

    const float4 ad4 = ((const float4*)a_dst)[n];
    const float adst[HEADS] = {ad4.x, ad4.y, ad4.z, ad4.w};

    // ---- pass 1: per-head max of leaky_relu(a_src[s] + a_dst[n]) ----
    float lmax[HEADS];
    #pragma unroll
    for (int i = 0; i < HEADS; ++i) lmax[i] = -INFINITY;
    for (int j = tid; j < deg; j += 256) {
        float4 as4 = a_src4[col_src[start + j]];
        float ev[HEADS] = {as4.x, as4.y, as4.z, as4.w};
        #pragma unroll
        for (int i = 0; i < HEADS; ++i) {
            float e = ev[i] + adst[i];
            e = (e > 0.f) ? e : NEG_SLOPE * e;
            lmax[i] = fmaxf(lmax[i], e);
        }
    }
    #pragma unroll
    for (int i = 0; i < HEADS; ++i) red[tid * HEADS + i] = lmax[i];
    __syncthreads();
    for (int s = 128; s > 0; s >>= 1) {
        if (tid < s)
            #pragma unroll
            for (int i = 0; i < HEADS; ++i)
                red[tid * HEADS + i] = fmaxf(red[tid * HEADS + i], red[(tid + s) * HEADS + i]);
        __syncthreads();
    }
    float emax[HEADS];
    #pragma unroll
    for (int i = 0; i < HEADS; ++i) emax[i] = red[i];
    __syncthreads();

    // ---- pass 2: per-head sum of exp(e - emax) ----
    float lsum[HEADS] = {0.f, 0.f, 0.f, 0.f};
    for (int j = tid; j < deg; j += 256) {
        float4 as4 = a_src4[col_src[start + j]];
        float ev[HEADS] = {as4.x, as4.y, as4.z, as4.w};
        #pragma unroll
        for (int i = 0; i < HEADS; ++i) {
            float e = ev[i] + adst[i];
            e = (e > 0.f) ? e : NEG_SLOPE * e;
            lsum[i] += expf(e - emax[i]);
        }
    }
    #pragma unroll
    for (int i = 0; i < HEADS; ++i) red[tid * HEADS + i] = lsum[i];
    __syncthreads();
    for (int s = 128; s > 0; s >>= 1) {
        if (tid < s)
            #pragma unroll
            for (int i = 0; i < HEADS; ++i) red[tid * HEADS + i] += red[(tid + s) * HEADS + i];
        __syncthreads();
    }
    float rden[HEADS];
    #pragma unroll
    for (int i = 0; i < HEADS; ++i) rden[i] = 1.0f / (red[i] + 1e-16f);
    __syncthreads();

    // ---- pass 3: accumulate alpha-weighted messages, chunked through LDS ----
    float acc[4] = {0.f, 0.f, 0.f, 0.f};   // channels 4*tid .. 4*tid+3 (head tid>>6)
    for (int base = 0; base < deg; base += CH) {
        int len = min(CH, deg - base);
        if (tid < len) {
            int s = col_src[start + base + tid];
            s_src[tid] = s;
            float4 as4 = a_src4[s];
            float ev[HEADS] = {as4.x, as4.y, as4.z, as4.w};
            #pragma unroll
            for (int i = 0; i < HEADS; ++i) {
                float e = ev[i] + adst[i];
                e = (e > 0.f) ? e : NEG_SLOPE * e;
                s_alpha[tid * HEADS + i] = expf(e - emax[i]) * rden[i];
            }
        }
        __syncthreads();
        for (int j = 0; j < len; ++j) {
            const float4 hv = *(const float4*)(hfeat + (size_t)s_src[j] * FDIM + tid * 4);
            const float a = s_alpha[j * HEADS + head];
            acc[0] += hv.x * a;
            acc[1] += hv.y * a;
            acc[2] += hv.z * a;
            acc[3] += hv.w * a;
        }
        __syncthreads();
    }

    // ---- head mean: red[hc] = value of global channel hc = head*256 + c ----
    #pragma unroll
    for (int k = 0; k < 4; ++k) red[tid * 4 + k] = acc[k];
    __syncthreads();
    float m = 0.25f * (red[tid] + red[256 + tid] + red[512 + tid] + red[768 + tid]) + bias[tid];
    if (do_elu) m = (m > 0.f) ? m : expm1f(m);
    out[(size_t)n * HID + tid] = m;
}

// ---------------------------------------------------------------------------
extern "C" void kernel_launch(void* const* d_in, const int* in_sizes, int n_in,
                              void* d_out, int out_size, void* d_ws, size_t ws_size,
                              hipStream_t stream) {
    const float*     x   = (const float*)d_in[0];
    const long long* ei  = (const long long*)d_in[1];   // int64 per reference
    const float*     W1  = (const float*)d_in[2];
    const float*     as1 = (const float*)d_in[3];
    const float*     ad1 = (const float*)d_in[4];
    const float*     b1  = (const float*)d_in[5];
    const float*     W2  = (const float*)d_in[6];
    const float*     as2 = (const float*)d_in[7];
    const float*     ad2 = (const float*)d_in[8];
    const float*     b2  = (const float*)d_in[9];
    float* out = (float*)d_out;

    const int E  = in_sizes[1] / 2;
    const int Et = E + N_NODES;

    // workspace carve-up (256B aligned)
    char* ws = (char*)d_ws;
    size_t off = 0;
    auto carve = [&](size_t bytes) -> void* {
        void* p = ws + off;
        off += (bytes + 255) & ~(size_t)255;
        return p;
    };
    float* hbuf    = (float*)carve((size_t)N_NODES * FDIM * 4);   // 41 MB (reused layer 2)
    float* out1    = (float*)carve((size_t)N_NODES * HID * 4);    // 10 MB
    float* a_src   = (float*)carve((size_t)N_NODES * HEADS * 4);
    float* a_dst   = (float*)carve((size_t)N_NODES * HEADS * 4);
    int*   row_ptr = (int*)carve((size_t)(N_NODES + 1) * 4);
    int*   cursor  = (int*)carve((size_t)N_NODES * 4);
    int*   counts  = (int*)carve((size_t)N_NODES * 4);
    int*   col_src = (int*)carve((size_t)Et * 4);

    // CSR build
    hipMemsetAsync(counts, 0, (size_t)N_NODES * 4, stream);
    count_kernel<<<(Et + 255) / 256, 256, 0, stream>>>(ei, E, counts);
    scan_kernel<<<1, 1024, 0, stream>>>(counts, row_ptr, cursor, N_NODES);
    fill_kernel<<<(Et + 255) / 256, 256, 0, stream>>>(ei, E, cursor, col_src);

    // ---- layer 1 ----
    gemm_wmma_f32<<<dim3(N_NODES / 16, FDIM / 64), 128, 16 * IN_DIM * 4, stream>>>(
        x, W1, hbuf, IN_DIM, FDIM);
    attn_kernel<<<N_NODES, 256, 0, stream>>>(hbuf, as1, ad1, a_src, a_dst);
    aggregate_kernel<<<N_NODES, 256, 0, stream>>>(hbuf, a_src, a_dst, row_ptr, col_src,
                                                  b1, out1, /*do_elu=*/1);

    // ---- layer 2 (hbuf reused for h2) ----
    gemm_wmma_f32<<<dim3(N_NODES / 16, FDIM / 64), 128, 16 * HID * 4, stream>>>(
        out1, W2, hbuf, HID, FDIM);
    attn_kernel<<<N_NODES, 256, 0, stream>>>(hbuf, as2, ad2, a_src, a_dst);
    aggregate_kernel<<<N_NODES, 256, 0, stream>>>(hbuf, a_src, a_dst, row_ptr, col_src,
                                                  b2, out, /*do_elu=*/0);
}